// GAT_2499670966351
// MI455X (gfx1250) — compile-verified
//
#include <hip/hip_runtime.h>
#include <math.h>

typedef __attribute__((ext_vector_type(2))) float v2f;
typedef __attribute__((ext_vector_type(8))) float v8f;

#define NEG_SLOPE 0.2f

// ---------------------------------------------------------------------------
// order-preserving float <-> uint key (for atomicMax-based segment max)
__device__ __forceinline__ unsigned f2key(float f) {
    unsigned b = __float_as_uint(f);
    return b ^ ((b >> 31) ? 0xFFFFFFFFu : 0x80000000u);
}
__device__ __forceinline__ float key2f(unsigned k) {
    return __uint_as_float((k & 0x80000000u) ? (k ^ 0x80000000u) : ~k);
}

// ---------------------------------------------------------------------------
// 1) per-dst degree + edge-attr sum (for self-loop fill_value='mean')
__global__ __launch_bounds__(256) void deg_kernel(
    const long long* __restrict__ ei, const float* __restrict__ ea,
    float* __restrict__ deg, float* __restrict__ attrsum, int E)
{
    int e = blockIdx.x * blockDim.x + threadIdx.x;
    if (e >= E) return;
    int d = (int)ei[(size_t)E + e];
    atomicAdd(&deg[d], 1.0f);
    atomicAdd(&attrsum[d], ea[e]);
}

__global__ __launch_bounds__(256) void loop_attr_kernel(
    const float* __restrict__ deg, const float* __restrict__ attrsum,
    float* __restrict__ loop_ea, int N)
{
    int i = blockIdx.x * blockDim.x + threadIdx.x;
    if (i >= N) return;
    loop_ea[i] = attrsum[i] / fmaxf(deg[i], 1.0f);
}

// ---------------------------------------------------------------------------
// 2) dual GEMM layer0: xl = x@Wl+bl, xr = x@Wr+br  (N x 128 @ 128 x 64)
//    f32 WMMA 16x16x4 -- exact fp32, one 16-row tile per block, 4 waves = 4 col tiles
__global__ __launch_bounds__(128) void gemm0_dual(
    const float* __restrict__ x,
    const float* __restrict__ Wl, const float* __restrict__ bl,
    const float* __restrict__ Wr, const float* __restrict__ br,
    float* __restrict__ xl, float* __restrict__ xr, int N)
{
    __shared__ float sx[16 * 132];          // 16 rows x 128, padded stride 132 (bank-conflict-free)
    const int r0  = blockIdx.x * 16;
    const int tid = threadIdx.x;

    if (r0 + 16 <= N) {
        // contiguous 2048-float block: fully coalesced
        #pragma unroll
        for (int i = 0; i < 16; ++i) {
            int j = tid + i * 128;
            sx[(j >> 7) * 132 + (j & 127)] = x[(size_t)r0 * 128 + j];
        }
    } else {
        #pragma unroll
        for (int i = 0; i < 16; ++i) {
            int j = tid + i * 128;
            int row = r0 + (j >> 7);
            sx[(j >> 7) * 132 + (j & 127)] = (row < N) ? x[(size_t)row * 128 + (j & 127)] : 0.f;
        }
    }
    __syncthreads();

    const int wave = tid >> 5;
    const int lane = tid & 31;
    const int m    = lane & 15;             // A row within tile (ISA 16x4 f32 A layout)
    const int half = lane >> 4;             // lanes 16-31 carry K+2,K+3
    const int n    = wave * 16 + (lane & 15); // output column

    v8f accL = {};
    v8f accR = {};
    for (int k = 0; k < 128; k += 4) {
        const int kb = k + half * 2;
        v2f a;  a.x  = sx[m * 132 + kb];               a.y  = sx[m * 132 + kb + 1];
        v2f bL; bL.x = Wl[(size_t)kb * 64 + n];        bL.y = Wl[(size_t)(kb + 1) * 64 + n];
        v2f bR; bR.x = Wr[(size_t)kb * 64 + n];        bR.y = Wr[(size_t)(kb + 1) * 64 + n];
        accL = __builtin_amdgcn_wmma_f32_16x16x4_f32(false, a, false, bL, (short)0, accL, false, false);
        accR = __builtin_amdgcn_wmma_f32_16x16x4_f32(false, a, false, bR, (short)0, accR, false, false);
    }

    const float bln = bl[n], brn = br[n];
    #pragma unroll
    for (int r = 0; r < 8; ++r) {
        int row = r0 + r + half * 8;        // ISA 16x16 f32 C/D layout
        if (row < N) {
            xl[(size_t)row * 64 + n] = accL[r] + bln;
            xr[(size_t)row * 64 + n] = accR[r] + brn;
        }
    }
}

// ---------------------------------------------------------------------------
// 3) layer0 edge logits: wave per edge, 64 features (2 per lane)
__global__ __launch_bounds__(256) void edge_logits0(
    const long long* __restrict__ ei, const float* __restrict__ ea,
    const float* __restrict__ loop_ea,
    const float* __restrict__ xl, const float* __restrict__ xr,
    const float* __restrict__ We, const float* __restrict__ att,
    float* __restrict__ logits, unsigned* __restrict__ maxkey,
    int E, int Etot)
{
    int w    = (blockIdx.x * blockDim.x + threadIdx.x) >> 5;
    int lane = threadIdx.x & 31;
    if (w >= Etot) return;
    int s, d; float a;
    if (w < E) { s = (int)ei[w]; d = (int)ei[(size_t)E + w]; a = ea[w]; }
    else       { s = d = w - E; a = loop_ea[s]; }

    float acc = 0.f;
    #pragma unroll
    for (int i = 0; i < 2; ++i) {
        int f = lane + 32 * i;
        float hv = xl[(size_t)s * 64 + f] + xr[(size_t)d * 64 + f] + a * We[f];
        hv = (hv > 0.f) ? hv : NEG_SLOPE * hv;
        acc += hv * att[f];
    }
    #pragma unroll
    for (int msk = 16; msk; msk >>= 1) acc += __shfl_xor(acc, msk, 32);

    if (lane == 0) {
        logits[w] = acc;
        atomicMax(&maxkey[d], f2key(acc));
    }
}

// 4) layer0 aggregation: num[d] += exp(l-m)*xl[s], den[d] += exp(l-m)
__global__ __launch_bounds__(256) void edge_agg0(
    const long long* __restrict__ ei,
    const float* __restrict__ xl,
    const float* __restrict__ logits, const unsigned* __restrict__ maxkey,
    float* __restrict__ den, float* __restrict__ num,
    int E, int Etot)
{
    int w    = (blockIdx.x * blockDim.x + threadIdx.x) >> 5;
    int lane = threadIdx.x & 31;
    if (w >= Etot) return;
    int s, d;
    if (w < E) { s = (int)ei[w]; d = (int)ei[(size_t)E + w]; }
    else       { s = d = w - E; }

    float p = __expf(logits[w] - key2f(maxkey[d]));
    if (lane == 0) atomicAdd(&den[d], p);
    #pragma unroll
    for (int i = 0; i < 2; ++i) {
        int f = lane + 32 * i;
        atomicAdd(&num[(size_t)d * 64 + f], p * xl[(size_t)s * 64 + f]);
    }
}

// 5) h0 = relu(num/den + bias) in place
__global__ __launch_bounds__(256) void finalize0(
    float* __restrict__ num, const float* __restrict__ den,
    const float* __restrict__ bias, int N)
{
    int idx = blockIdx.x * blockDim.x + threadIdx.x;
    if (idx >= N * 64) return;
    float h = num[idx] / (den[idx >> 6] + 1e-16f) + bias[idx & 63];
    num[idx] = (h > 0.f) ? h : 0.f;
}

// ---------------------------------------------------------------------------
// 6) dual GEMM layer1: (N x 64) @ (64 x 10->16 padded), W staged zero-padded in LDS
__global__ __launch_bounds__(128) void gemm1_dual(
    const float* __restrict__ h,
    const float* __restrict__ Wl, const float* __restrict__ bl,
    const float* __restrict__ Wr, const float* __restrict__ br,
    float* __restrict__ xl, float* __restrict__ xr,
    int N, int ntiles)
{
    __shared__ float sWl[64 * 16];
    __shared__ float sWr[64 * 16];
    const int tid = threadIdx.x;
    for (int j = tid; j < 64 * 16; j += 128) {
        int kk = j >> 4, nn = j & 15;
        sWl[j] = (nn < 10) ? Wl[kk * 10 + nn] : 0.f;
        sWr[j] = (nn < 10) ? Wr[kk * 10 + nn] : 0.f;
    }
    __syncthreads();

    const int wave = tid >> 5, lane = tid & 31;
    const int tile = blockIdx.x * 4 + wave;
    if (tile >= ntiles) return;
    const int r0   = tile * 16;
    const int m    = lane & 15;
    const int half = lane >> 4;
    const int n    = lane & 15;
    int arow = r0 + m; if (arow >= N) arow = N - 1;

    v8f accL = {};
    v8f accR = {};
    for (int k = 0; k < 64; k += 4) {
        const int kb = k + half * 2;
        v2f a;  a.x  = h[(size_t)arow * 64 + kb];  a.y  = h[(size_t)arow * 64 + kb + 1];
        v2f bL; bL.x = sWl[kb * 16 + n];           bL.y = sWl[(kb + 1) * 16 + n];
        v2f bR; bR.x = sWr[kb * 16 + n];           bR.y = sWr[(kb + 1) * 16 + n];
        accL = __builtin_amdgcn_wmma_f32_16x16x4_f32(false, a, false, bL, (short)0, accL, false, false);
        accR = __builtin_amdgcn_wmma_f32_16x16x4_f32(false, a, false, bR, (short)0, accR, false, false);
    }

    const float bln = (n < 10) ? bl[n] : 0.f;
    const float brn = (n < 10) ? br[n] : 0.f;
    #pragma unroll
    for (int r = 0; r < 8; ++r) {
        int row = r0 + r + half * 8;
        if (row < N) {
            xl[(size_t)row * 16 + n] = accL[r] + bln;   // stride 16 (cols 10..15 = 0)
            xr[(size_t)row * 16 + n] = accR[r] + brn;
        }
    }
}

// 7) layer1 edge logits: thread per edge (10 features)
__global__ __launch_bounds__(256) void edge_logits1(
    const long long* __restrict__ ei, const float* __restrict__ ea,
    const float* __restrict__ loop_ea,
    const float* __restrict__ xl, const float* __restrict__ xr,
    const float* __restrict__ We, const float* __restrict__ att,
    float* __restrict__ logits, unsigned* __restrict__ maxkey,
    int E, int Etot)
{
    int w = blockIdx.x * blockDim.x + threadIdx.x;
    if (w >= Etot) return;
    int s, d; float a;
    if (w < E) { s = (int)ei[w]; d = (int)ei[(size_t)E + w]; a = ea[w]; }
    else       { s = d = w - E; a = loop_ea[s]; }

    float acc = 0.f;
    #pragma unroll
    for (int c = 0; c < 10; ++c) {
        float hv = xl[(size_t)s * 16 + c] + xr[(size_t)d * 16 + c] + a * We[c];
        hv = (hv > 0.f) ? hv : NEG_SLOPE * hv;
        acc += hv * att[c];
    }
    logits[w] = acc;
    atomicMax(&maxkey[d], f2key(acc));
}

// 8) layer1 aggregation
__global__ __launch_bounds__(256) void edge_agg1(
    const long long* __restrict__ ei,
    const float* __restrict__ xl,
    const float* __restrict__ logits, const unsigned* __restrict__ maxkey,
    float* __restrict__ den, float* __restrict__ num,
    int E, int Etot)
{
    int w = blockIdx.x * blockDim.x + threadIdx.x;
    if (w >= Etot) return;
    int s, d;
    if (w < E) { s = (int)ei[w]; d = (int)ei[(size_t)E + w]; }
    else       { s = d = w - E; }

    float p = __expf(logits[w] - key2f(maxkey[d]));
    atomicAdd(&den[d], p);
    #pragma unroll
    for (int c = 0; c < 10; ++c)
        atomicAdd(&num[(size_t)d * 16 + c], p * xl[(size_t)s * 16 + c]);
}

// 9) out = log_softmax(num/den + bias)
__global__ __launch_bounds__(256) void finalize1(
    const float* __restrict__ num, const float* __restrict__ den,
    const float* __restrict__ bias, float* __restrict__ out, int N)
{
    int i = blockIdx.x * blockDim.x + threadIdx.x;
    if (i >= N) return;
    float inv = 1.0f / (den[i] + 1e-16f);
    float v[10], mx = -INFINITY;
    #pragma unroll
    for (int c = 0; c < 10; ++c) {
        v[c] = num[(size_t)i * 16 + c] * inv + bias[c];
        mx = fmaxf(mx, v[c]);
    }
    float s = 0.f;
    #pragma unroll
    for (int c = 0; c < 10; ++c) s += __expf(v[c] - mx);
    float lse = mx + __logf(s);
    #pragma unroll
    for (int c = 0; c < 10; ++c) out[(size_t)i * 10 + c] = v[c] - lse;
}

// ---------------------------------------------------------------------------
extern "C" void kernel_launch(void* const* d_in, const int* in_sizes, int n_in,
                              void* d_out, int out_size, void* d_ws, size_t ws_size,
                              hipStream_t stream) {
    const float*     x     = (const float*)d_in[0];
    const long long* ei    = (const long long*)d_in[1];   // int64 (2, E)
    const float*     ea    = (const float*)d_in[2];
    const float*     Wl0   = (const float*)d_in[3];
    const float*     bl0   = (const float*)d_in[4];
    const float*     Wr0   = (const float*)d_in[5];
    const float*     br0   = (const float*)d_in[6];
    const float*     We0   = (const float*)d_in[7];
    const float*     att0  = (const float*)d_in[8];
    const float*     bias0 = (const float*)d_in[9];
    const float*     Wl1   = (const float*)d_in[10];
    const float*     bl1   = (const float*)d_in[11];
    const float*     Wr1   = (const float*)d_in[12];
    const float*     br1   = (const float*)d_in[13];
    const float*     We1   = (const float*)d_in[14];
    const float*     att1  = (const float*)d_in[15];
    const float*     bias1 = (const float*)d_in[16];
    float* out = (float*)d_out;

    const int N    = in_sizes[0] / 128;
    const int E    = in_sizes[2];
    const int Etot = E + N;

    // workspace carve-up (floats)
    float* ws = (float*)d_ws;
    size_t o = 0;
    float*    deg     = ws + o; o += N;
    float*    attrsum = ws + o; o += N;               // [deg|attrsum] zeroed together
    float*    loop_ea = ws + o; o += N;
    float*    xl0     = ws + o; o += (size_t)N * 64;
    float*    xr0     = ws + o; o += (size_t)N * 64;
    float*    num0    = ws + o; o += (size_t)N * 64;  // [num0|den0|key0] zeroed together
    float*    den0    = ws + o; o += N;
    unsigned* key0    = (unsigned*)(ws + o); o += N;
    float*    xl1     = ws + o; o += (size_t)N * 16;
    float*    xr1     = ws + o; o += (size_t)N * 16;
    float*    num1    = ws + o; o += (size_t)N * 16;  // [num1|den1|key1] zeroed together
    float*    den1    = ws + o; o += N;
    unsigned* key1    = (unsigned*)(ws + o); o += N;
    float*    logits  = ws + o; o += Etot;            // reused for both layers

    // zero accumulators (graph-capture-safe, every call)
    hipMemsetAsync(deg,  0, sizeof(float) * 2 * (size_t)N, stream);
    hipMemsetAsync(num0, 0, sizeof(float) * ((size_t)N * 64 + 2 * (size_t)N), stream);
    hipMemsetAsync(num1, 0, sizeof(float) * ((size_t)N * 16 + 2 * (size_t)N), stream);

    const int rowTiles = (N + 15) / 16;

    // self-loop mean attr
    deg_kernel<<<(E + 255) / 256, 256, 0, stream>>>(ei, ea, deg, attrsum, E);
    loop_attr_kernel<<<(N + 255) / 256, 256, 0, stream>>>(deg, attrsum, loop_ea, N);

    // layer 0
    gemm0_dual<<<rowTiles, 128, 0, stream>>>(x, Wl0, bl0, Wr0, br0, xl0, xr0, N);
    edge_logits0<<<(Etot + 7) / 8, 256, 0, stream>>>(ei, ea, loop_ea, xl0, xr0, We0, att0,
                                                     logits, key0, E, Etot);
    edge_agg0<<<(Etot + 7) / 8, 256, 0, stream>>>(ei, xl0, logits, key0, den0, num0, E, Etot);
    finalize0<<<((size_t)N * 64 + 255) / 256, 256, 0, stream>>>(num0, den0, bias0, N);

    // layer 1 (h0 lives in num0)
    gemm1_dual<<<(rowTiles + 3) / 4, 128, 0, stream>>>(num0, Wl1, bl1, Wr1, br1, xl1, xr1,
                                                       N, rowTiles);
    edge_logits1<<<(Etot + 255) / 256, 256, 0, stream>>>(ei, ea, loop_ea, xl1, xr1, We1, att1,
                                                         logits, key1, E, Etot);
    edge_agg1<<<(Etot + 255) / 256, 256, 0, stream>>>(ei, xl1, logits, key1, den1, num1, E, Etot);
    finalize1<<<(N + 255) / 256, 256, 0, stream>>>(num1, den1, bias1, out, N);
}